// TextEmbedderLETS_8169027797036
// MI455X (gfx1250) — compile-verified
//
#include <hip/hip_runtime.h>
#include <hip/hip_bf16.h>

// Problem constants (from reference setup_inputs)
#define Bq 8
#define Cq 7
#define Tq 45
#define Sq 4096
#define Dq 5120
#define KPAD 48          // T padded to multiple of 4 for WMMA f32 16x16x4
#define MPAD 16          // batch rows padded to WMMA M
#define SCHUNKS 32       // S split for the streaming mean-pool
#define SPER (Sq / SCHUNKS)   // 128 rows per chunk

typedef __attribute__((ext_vector_type(2))) float v2f;
typedef __attribute__((ext_vector_type(4))) float v4f;
typedef __attribute__((ext_vector_type(8))) float v8f;

__device__ __forceinline__ float wave_sum(float v) {
    #pragma unroll
    for (int off = 16; off >= 1; off >>= 1) v += __shfl_xor(v, off, 32);
    return v;
}
__device__ __forceinline__ float wave_min(float v) {
    #pragma unroll
    for (int off = 16; off >= 1; off >>= 1) v = fminf(v, __shfl_xor(v, off, 32));
    return v;
}
__device__ __forceinline__ float wave_max(float v) {
    #pragma unroll
    for (int off = 16; off >= 1; off >>= 1) v = fmaxf(v, __shfl_xor(v, off, 32));
    return v;
}

// ---------------------------------------------------------------------------
// K1: streaming partial mean-pool of hidden_states (the 671 MB stage).
// grid (Dq/1024, SCHUNKS, Bq), block 256. Each thread: 4-float column strip,
// nontemporal b128 loads (stream-once data), deterministic partial sums.
// part layout: [SCHUNKS][Bq][Dq]
// ---------------------------------------------------------------------------
__global__ void pool_partial_k(const float* __restrict__ H, float* __restrict__ part) {
    const int d0 = blockIdx.x * 1024 + threadIdx.x * 4;
    const int sc = blockIdx.y;
    const int b  = blockIdx.z;
    const v4f* p = (const v4f*)(H + ((size_t)b * Sq + (size_t)sc * SPER) * Dq + d0);
    v4f acc = {0.f, 0.f, 0.f, 0.f};
    #pragma unroll 4
    for (int s = 0; s < SPER; ++s) {
        v4f v = __builtin_nontemporal_load(p);
        p += Dq / 4;
        acc += v;
    }
    v4f* o = (v4f*)(part + ((size_t)sc * Bq + b) * Dq + d0);
    *o = acc;
}

// ---------------------------------------------------------------------------
// K2: per-(b,c) min-max normalize over T. grid B*C blocks of 1 wave.
// Writes x_normalized at d_out + Bq*Dq.
// ---------------------------------------------------------------------------
__global__ void minmax_norm_k(const float* __restrict__ x, float* __restrict__ xn) {
    const int bc = blockIdx.x;                    // 0..55
    const float* row = x + (size_t)bc * Tq;
    float*       orow = xn + (size_t)bc * Tq;
    const int t = threadIdx.x;                    // 0..31
    const bool a_ok = (t < Tq), b_ok = (t + 32 < Tq);
    const float v1 = a_ok ? row[t]      : 0.f;
    const float v2 = b_ok ? row[t + 32] : 0.f;
    float mn = fminf(a_ok ? v1 :  3.402823466e38f, b_ok ? v2 :  3.402823466e38f);
    float mx = fmaxf(a_ok ? v1 : -3.402823466e38f, b_ok ? v2 : -3.402823466e38f);
    mn = wave_min(mn);
    mx = wave_max(mx);
    const bool  pass  = !(mx > mn);
    const float scale = pass ? 1.f : 1.f / (mx - mn);
    if (a_ok) orow[t]      = pass ? v1 : (v1 - mn) * scale;
    if (b_ok) orow[t + 32] = pass ? v2 : (v2 - mn) * scale;
}

// ---------------------------------------------------------------------------
// K3: padded series_mean: sm[MPAD][KPAD]; rows >= Bq and cols >= Tq are zero
// so the WMMA GEMM needs no predication at all.
// ---------------------------------------------------------------------------
__global__ void series_mean_k(const float* __restrict__ x, float* __restrict__ sm) {
    const int b = blockIdx.x;                     // 0..15
    const int t = threadIdx.x;                    // 64 threads
    if (t < KPAD) {
        float s = 0.f;
        if (b < Bq && t < Tq) {
            #pragma unroll
            for (int c = 0; c < Cq; ++c) s += x[((size_t)b * Cq + c) * Tq + t];
            s *= (1.f / (float)Cq);
        }
        sm[b * KPAD + t] = s;
    }
}

// ---------------------------------------------------------------------------
// K3b: zero-pad W_proj [45][5120] -> Wpad [48][5120] (rows 45..47 = 0) so the
// GEMM inner loop is branch-free (no exec-mask diamonds around loads).
// ---------------------------------------------------------------------------
__global__ void pad_w_k(const float* __restrict__ W, float* __restrict__ Wp) {
    const int idx = blockIdx.x * 256 + threadIdx.x;   // < KPAD*Dq
    const int k = idx / Dq;
    Wp[idx] = (k < Tq) ? W[idx] : 0.f;
}

// ---------------------------------------------------------------------------
// K4: GEMM emb[16 x 5120] = sm[16 x 48] @ Wpad[48 x 5120] + bias
// via V_WMMA_F32_16X16X4_F32 (full fp32 matrix pipe, 12 K-steps per tile).
// grid 320 blocks x 1 wave; each wave owns one 16x16 N-tile. All operands are
// pre-padded -> every load unconditional, EXEC all-1s throughout.
// A layout (ISA 7.12.2, 32-bit A 16x4): lane l -> M=l&15, k = k0+((l>>4)<<1)+{0,1}
//   -> contiguous v2f per lane per step (b64 load).
// B layout (4x16, rows striped across lanes per VGPR): same kb, N=n0+(l&15).
// C/D: VGPR v, lanes 0-15 -> M=v (valid batches 0..7), lanes 16-31 -> M=v+8 (pad)
// ---------------------------------------------------------------------------
__global__ void gemm_wmma_k(const float* __restrict__ sm,
                            const float* __restrict__ Wp,
                            const float* __restrict__ bias,
                            float* __restrict__ emb) {
    const int n0   = blockIdx.x * 16;
    const int l    = threadIdx.x;        // 0..31
    const int half = l >> 4;             // 0|1
    const int m    = l & 15;
    const int n    = n0 + m;

    const float* Arow = sm + m * KPAD;          // zero rows for m >= 8
    const float* Bcol = Wp + n;                 // zero rows for k >= 45

#if __has_builtin(__builtin_amdgcn_wmma_f32_16x16x4_f32)
    v8f c = {};
    #pragma unroll
    for (int k0 = 0; k0 < KPAD; k0 += 4) {
        const int kb = k0 + (half << 1);
        v2f a = *(const v2f*)(Arow + kb);        // A[m][kb], A[m][kb+1]
        v2f bm;
        bm.x = Bcol[(size_t)kb * Dq];            // W[kb][n]
        bm.y = Bcol[(size_t)(kb + 1) * Dq];      // W[kb+1][n]
        c = __builtin_amdgcn_wmma_f32_16x16x4_f32(
                /*neg_a=*/false, a, /*neg_b=*/false, bm,
                /*c_mod=*/(short)0, c, /*reuse_a=*/false, /*reuse_b=*/false);
    }
    if (l < 16) {                        // lanes 0..15 hold M=0..7 in VGPRs 0..7
        const float bb = bias[n];
        #pragma unroll
        for (int v = 0; v < Bq; ++v)
            emb[(size_t)v * Dq + n] = c[v] + bb;
    }
#else
    // Scalar fallback (keeps compile green if builtin absent on this toolchain)
    if (l < 16) {
        const float bb = bias[n];
        #pragma unroll
        for (int v = 0; v < Bq; ++v) {
            float acc = bb;
            for (int k = 0; k < Tq; ++k)
                acc = fmaf(sm[v * KPAD + k], Wp[(size_t)k * Dq + n], acc);
            emb[(size_t)v * Dq + n] = acc;
        }
    }
#endif
}

// ---------------------------------------------------------------------------
// K5: per-batch inverse L2 norm of emb (fixed-order tree reduction -> deterministic)
// ---------------------------------------------------------------------------
__global__ void inv_norm_k(const float* __restrict__ emb, float* __restrict__ invn) {
    const int b = blockIdx.x;
    float s = 0.f;
    for (int d = threadIdx.x; d < Dq; d += 256) {
        const float v = emb[(size_t)b * Dq + d];
        s = fmaf(v, v, s);
    }
    s = wave_sum(s);
    __shared__ float sh[8];
    const int lane = threadIdx.x & 31, w = threadIdx.x >> 5;
    if (lane == 0) sh[w] = s;
    __syncthreads();
    if (w == 0) {
        float t = (lane < 8) ? sh[lane] : 0.f;
        t = wave_sum(t);
        if (lane == 0) invn[b] = 1.f / fmaxf(sqrtf(t), 1e-12f);
    }
}

// ---------------------------------------------------------------------------
// K6: e_fused[b][d] = (sum_k part[k][b][d]) / S + emb[b][d] * invn[b]
// ---------------------------------------------------------------------------
__global__ void fuse_k(const float* __restrict__ part, const float* __restrict__ emb,
                       const float* __restrict__ invn, float* __restrict__ out) {
    const int idx = blockIdx.x * 256 + threadIdx.x;   // < 40960
    const int b = idx / Dq;
    const int d = idx - b * Dq;
    float acc = 0.f;
    #pragma unroll
    for (int k = 0; k < SCHUNKS; ++k)
        acc += part[((size_t)k * Bq + b) * Dq + d];
    out[idx] = acc * (1.f / (float)Sq) + emb[idx] * invn[b];
}

extern "C" void kernel_launch(void* const* d_in, const int* in_sizes, int n_in,
                              void* d_out, int out_size, void* d_ws, size_t ws_size,
                              hipStream_t stream) {
    const float* x    = (const float*)d_in[0];   // [8,7,45]
    const float* H    = (const float*)d_in[1];   // [8,4096,5120]
    const float* W    = (const float*)d_in[2];   // [45,5120]
    const float* bias = (const float*)d_in[3];   // [5120]
    float* out = (float*)d_out;                  // [8*5120] e_fused ++ [8*7*45] x_norm

    float* ws   = (float*)d_ws;
    float* part = ws;                               // SCHUNKS*Bq*Dq = 1,310,720 f
    float* emb  = part + (size_t)SCHUNKS * Bq * Dq; // Bq*Dq  = 40,960 f
    float* Wp   = emb  + (size_t)Bq * Dq;           // KPAD*Dq = 245,760 f
    float* sm   = Wp   + (size_t)KPAD * Dq;         // MPAD*KPAD = 768 f
    float* invn = sm   + MPAD * KPAD;               // 8 f

    // Big streaming stage: 1280 blocks x 256 threads, b128 NT loads
    pool_partial_k<<<dim3(Dq / 1024, SCHUNKS, Bq), 256, 0, stream>>>(H, part);
    // Small stages
    minmax_norm_k<<<Bq * Cq, 32, 0, stream>>>(x, out + Bq * Dq);
    series_mean_k<<<MPAD, 64, 0, stream>>>(x, sm);
    pad_w_k<<<(KPAD * Dq) / 256, 256, 0, stream>>>(W, Wp);
    gemm_wmma_k<<<Dq / 16, 32, 0, stream>>>(sm, Wp, bias, emb);
    inv_norm_k<<<Bq, 256, 0, stream>>>(emb, invn);
    fuse_k<<<(Bq * Dq) / 256, 256, 0, stream>>>(part, emb, invn, out);
}